// Forward_Model_27616639714074
// MI455X (gfx1250) — compile-verified
//
#include <hip/hip_runtime.h>

// Elementwise streaming kernel: 2-link IK angles per (x,y) point.
// Memory-bound: 128 MB traffic @ 23.3 TB/s ~ 5.5 us floor. No matmul
// structure -> no WMMA; optimize data path (NT b128 loads/stores, scalar
// uniform weight math) and keep VALU below the HBM floor (fast rcp divide,
// cos/sin of acos folded algebraically).

typedef __attribute__((ext_vector_type(4))) float fv4;

#define EPSC 1e-6f

constexpr int THREADS = 256;   // 8 wave32s per block
constexpr int NPER    = 4;     // float4s per thread -> 64B load / 64B store per thread

__device__ __forceinline__ float clamp1(float a) {
    return fminf(fmaxf(a, -1.0f), 1.0f);
}

// Fast divide: v_rcp_f32 (1 ulp) + one Newton step -> ~0.5 ulp.
// Safe here: denominators are >= 1e-6 and finite (sum of squares + EPS).
__device__ __forceinline__ float fast_div(float n, float d) {
    float r = __builtin_amdgcn_rcpf(d);
    r = fmaf(r, fmaf(-d, r, 1.0f), r);
    return n * r;
}

// Solve one point. Uses exact identities cos(acos(g))=g, sin(acos(g))=sqrt(1-g^2)
// (acos range is [0,pi] so sin >= 0).
__device__ __forceinline__ void solve_pt(float x, float y,
                                         float c0, float c1,
                                         float l1s, float l2s,
                                         float ll, float inv_den,
                                         float& t1a, float& t2a) {
    float xs  = x * c0;
    float ys  = y * c1;
    float dis = xs * xs + ys * ys + EPSC;
    float arg = clamp1((dis - ll) * inv_den);
    t2a = acosf(arg);
    float st  = sqrtf(fmaxf(1.0f - arg * arg, 0.0f));     // sin(t2a), exact math
    float num = ys * (l1s + l2s * arg) - xs * (l2s * st); // l2s*cos(t2a) == l2s*arg
    t1a = asinf(clamp1(fast_div(num, dis)));
}

__global__ __launch_bounds__(THREADS)
void ik_forward_kernel(const fv4* __restrict__ in,
                       const float* __restrict__ w,
                       fv4* __restrict__ out,
                       int n4) {
    // Per-mult scalars: uniform address -> scalar loads + SALU.
    // c = max(round_half_even(w), 0.001) — matches jnp.round + clip lower bound.
    float c0 = fmaxf(rintf(w[0]), 0.001f);
    float c1 = fmaxf(rintf(w[1]), 0.001f);
    float c2 = fmaxf(rintf(w[2]), 0.001f);
    float c3 = fmaxf(rintf(w[3]), 0.001f);
    float l1s = 0.5f * c2;
    float l2s = 0.5f * c3;
    float ll      = l1s * l1s + l2s * l2s;
    float inv_den = fast_div(1.0f, fmaf(2.0f * l1s, l2s, EPSC));

    int base = blockIdx.x * (THREADS * NPER) + threadIdx.x;

    if (base + (NPER - 1) * THREADS < n4) {
        // Fast path: issue all 4 non-temporal b128 loads back-to-back (MLP),
        // then compute, then 4 non-temporal b128 stores.
        fv4 v[NPER];
        #pragma unroll
        for (int k = 0; k < NPER; ++k)
            v[k] = __builtin_nontemporal_load(&in[base + k * THREADS]);

        fv4 o[NPER];
        #pragma unroll
        for (int k = 0; k < NPER; ++k) {
            float t1, t2;
            solve_pt(v[k].x, v[k].y, c0, c1, l1s, l2s, ll, inv_den, t1, t2);
            o[k].x = t1; o[k].y = t2;
            solve_pt(v[k].z, v[k].w, c0, c1, l1s, l2s, ll, inv_den, t1, t2);
            o[k].z = t1; o[k].w = t2;
        }

        #pragma unroll
        for (int k = 0; k < NPER; ++k)
            __builtin_nontemporal_store(o[k], &out[base + k * THREADS]);
    } else {
        // Guarded tail (not taken for SIZE=8388608, kept for generality).
        for (int k = 0; k < NPER; ++k) {
            int idx = base + k * THREADS;
            if (idx < n4) {
                fv4 v = __builtin_nontemporal_load(&in[idx]);
                fv4 o;
                float t1, t2;
                solve_pt(v.x, v.y, c0, c1, l1s, l2s, ll, inv_den, t1, t2);
                o.x = t1; o.y = t2;
                solve_pt(v.z, v.w, c0, c1, l1s, l2s, ll, inv_den, t1, t2);
                o.z = t1; o.w = t2;
                __builtin_nontemporal_store(o, &out[idx]);
            }
        }
    }
}

extern "C" void kernel_launch(void* const* d_in, const int* in_sizes, int n_in,
                              void* d_out, int out_size, void* d_ws, size_t ws_size,
                              hipStream_t stream) {
    (void)n_in; (void)d_ws; (void)ws_size;

    const fv4*   in  = (const fv4*)d_in[0];    // (1, SIZE, 2) float32, interleaved x,y
    const float* w   = (const float*)d_in[1];  // (N_MULTS, 4) float32
    fv4*         out = (fv4*)d_out;            // (N_MULTS, 1, SIZE, 2) float32

    const int n4      = in_sizes[0] / 4;       // float4 count = SIZE*2/4
    const int n_mults = in_sizes[1] / 4;       // N_MULTS (== 1 in reference)

    const int per_block = THREADS * NPER;
    const int blocks    = (n4 + per_block - 1) / per_block;

    for (int m = 0; m < n_mults; ++m) {
        ik_forward_kernel<<<blocks, THREADS, 0, stream>>>(
            in, w + 4 * m, out + (size_t)m * n4, n4);
    }
}